// GPRMamba2_74680891343688
// MI455X (gfx1250) — compile-verified
//
#include <hip/hip_runtime.h>
#include <math.h>

#define B0 8
#define CIN 3
#define HIMG 340
#define WIMG 720
#define GH 17
#define GW 90
#define PH 20
#define PW 8
#define HID 128
#define NL 4
#define NH 4
#define HD 64
#define NSTATE 128
#define DIN 256      // NH*HD
#define CONVD 512    // DIN + 2*NSTATE
#define PROJ 772     // DIN + CONVD + NH
#define KCONV 4
#define LSEQ 1530    // GH*GW
#define BT 16        // 2*B0 (fwd + reversed)
#define EPSF 1e-5f

typedef float v2f __attribute__((ext_vector_type(2)));
typedef float v8f __attribute__((ext_vector_type(8)));
typedef int   v4i __attribute__((ext_vector_type(4)));

typedef __attribute__((address_space(1))) v4i* gbl_v4i_p;   // global int4*
typedef __attribute__((address_space(3))) v4i* lds_v4i_p;   // LDS int4*

#if defined(__has_builtin)
#if __has_builtin(__builtin_amdgcn_global_load_async_to_lds_b128) && \
    __has_builtin(__builtin_amdgcn_s_wait_asynccnt)
#define HAVE_ASYNC_LDS 1
#endif
#endif

// ---------------------------------------------------------------------------
// Stage a 16 x KD fp32 A-tile (row stride = KD) from global into LDS,
// cooperatively across 128 threads.  Uses the gfx1250 async global->LDS
// path (ASYNCcnt) when available, plain loads otherwise.  Ends with barrier.
// ---------------------------------------------------------------------------
template <int KD>
__device__ __forceinline__ void stage_tile(const float* __restrict__ src,
                                           float* lds, int tid) {
    constexpr int ROUNDS = (16 * KD) / (128 * 4); // 4 floats (16B) per thread/round
#pragma unroll
    for (int r = 0; r < ROUNDS; ++r) {
        int f = r * 512 + tid * 4;          // linear float index in tile
        int row = f / KD, k = f % KD;       // KD is a power of two
#ifdef HAVE_ASYNC_LDS
        __builtin_amdgcn_global_load_async_to_lds_b128(
            (gbl_v4i_p)(src + (size_t)row * KD + k),
            (lds_v4i_p)(lds + f),
            0, 0);
#else
        float4 v = *(const float4*)(src + (size_t)row * KD + k);
        *(float4*)(lds + f) = v;
#endif
    }
#ifdef HAVE_ASYNC_LDS
    __builtin_amdgcn_s_wait_asynccnt(0);
#endif
    __syncthreads();
}

// ---------------------------------------------------------------------------
// fp32 WMMA 16x16 tile:  acc += A(16xK) * W(NxK)^T  for one 16-col tile.
// Arow: pointer to A row (rowbase + lane&15), stride-1 in K (LDS or global).
// Wrow: pointer to W row (col = tilebase + lane&15), stride-1 in K.
// hf = lane>>4 selects which K-pair this lane half feeds.
// ---------------------------------------------------------------------------
__device__ __forceinline__ v8f wmma_tile_f32(const float* __restrict__ Arow,
                                             const float* __restrict__ Wrow,
                                             int K, int hf) {
    v8f acc;
#pragma unroll
    for (int r = 0; r < 8; ++r) acc[r] = 0.f;
    for (int k0 = 0; k0 < K; k0 += 4) {
        int k = k0 + 2 * hf;
        v2f a; a.x = Arow[k]; a.y = Arow[k + 1];
        v2f b; b.x = Wrow[k]; b.y = Wrow[k + 1];
        acc = __builtin_amdgcn_wmma_f32_16x16x4_f32(false, a, false, b,
                                                    (short)0, acc, false, false);
    }
    return acc;
}

__device__ __forceinline__ float wave_sum32(float v) {
#pragma unroll
    for (int off = 16; off > 0; off >>= 1) v += __shfl_xor(v, off, 32);
    return v;
}

__device__ __forceinline__ float siluf(float x) { return x / (1.f + expf(-x)); }

// ---------------------------------------------------------------------------
// Patch embedding + gain channel; writes forward sequence h[b] and reversed
// copy h[b+8].  One 128-thread block per (b, grid position).
// ---------------------------------------------------------------------------
__global__ void patch_kernel(const float* __restrict__ x,
                             const float* __restrict__ pw,
                             const float* __restrict__ pb,
                             float* __restrict__ h) {
    __shared__ float sp[CIN * PH * PW]; // 480 patch values
    int bl = blockIdx.x;                 // b*LSEQ + l
    int b = bl / LSEQ, l = bl % LSEQ;
    int i = l / GW, j = l % GW;
    for (int idx = threadIdx.x; idx < CIN * PH * PW; idx += 128) {
        int c = idx / (PH * PW);
        int rem = idx % (PH * PW);
        int p = rem >> 3, q = rem & 7;
        sp[idx] = x[(((size_t)b * CIN + c) * HIMG + i * PH + p) * WIMG + j * PW + q];
    }
    __syncthreads();
    int o = threadIdx.x;
    float acc;
    if (o < HID - 1) {
        acc = pb[o];
        const float* wr = pw + (size_t)o * (CIN * PH * PW);
        for (int t = 0; t < CIN * PH * PW; ++t) acc += sp[t] * wr[t];
    } else {
        acc = (float)i / (float)(GH - 1); // gain channel
    }
    h[((size_t)b * LSEQ + l) * HID + o] = acc;
    h[((size_t)(b + B0) * LSEQ + (LSEQ - 1 - l)) * HID + o] = acc;
}

// ---------------------------------------------------------------------------
// RMS norm over 128 channels, one wave32 per row (4 values per lane).
// ---------------------------------------------------------------------------
__global__ void rms_kernel(const float* __restrict__ in,
                           const float* __restrict__ w,
                           float* __restrict__ out, int rows) {
    int row = blockIdx.x * 8 + (threadIdx.x >> 5);
    if (row >= rows) return;
    int lane = threadIdx.x & 31;
    const float4* p = (const float4*)(in + (size_t)row * HID);
    float4 v = p[lane];
    float ss = wave_sum32(v.x * v.x + v.y * v.y + v.z * v.z + v.w * v.w);
    float sc = rsqrtf(ss / (float)HID + EPSF);
    float4 wv = ((const float4*)w)[lane];
    float4 o;
    o.x = v.x * sc * wv.x; o.y = v.y * sc * wv.y;
    o.z = v.z * sc * wv.z; o.w = v.w * sc * wv.w;
    ((float4*)(out + (size_t)row * HID))[lane] = o;
}

// ---------------------------------------------------------------------------
// In-projection GEMM:  P[m,n] = U[m,:] . in_w[n,:] + in_b[n]
// M = BT*LSEQ, K = 128, N = 772 (49 col tiles, last partial).
// A-tile staged to LDS (async), 4 wave32 per block, one 16x16 tile per wave.
// ---------------------------------------------------------------------------
__global__ void inproj_kernel(const float* __restrict__ U,
                              const float* __restrict__ W,
                              const float* __restrict__ bias,
                              float* __restrict__ P) {
    __shared__ float lA[16 * HID];
    int wave = threadIdx.x >> 5, lane = threadIdx.x & 31;
    int hf = lane >> 4, ln = lane & 15;
    int rowb = blockIdx.x * 16;
    stage_tile<HID>(U + (size_t)rowb * HID, lA, threadIdx.x);

    int ntile = blockIdx.y * 4 + wave;
    if (ntile < (PROJ + 15) / 16) {              // wave-uniform
        int ncol = ntile * 16 + ln;
        bool cval = ncol < PROJ;
        const float* Arow = lA + ln * HID;
        const float* Wrow = W + (size_t)(cval ? ncol : 0) * HID;
        v8f acc = wmma_tile_f32(Arow, Wrow, HID, hf);
        if (cval) {
            float bv = bias[ncol];
#pragma unroll
            for (int r = 0; r < 8; ++r) {
                int row = rowb + r + 8 * hf;
                P[(size_t)row * PROJ + ncol] = acc[r] + bv;
            }
        }
    }
}

// ---------------------------------------------------------------------------
// Depthwise causal conv (K=4) + SiLU over the xBC slice of the projection.
// ---------------------------------------------------------------------------
__global__ void conv_kernel(const float* __restrict__ P,
                            const float* __restrict__ cw,
                            const float* __restrict__ cb,
                            float* __restrict__ XC) {
    size_t idx = (size_t)blockIdx.x * blockDim.x + threadIdx.x;
    size_t total = (size_t)BT * LSEQ * CONVD;
    if (idx >= total) return;
    int c = (int)(idx % CONVD);
    size_t bl = idx / CONVD;
    int l = (int)(bl % LSEQ);
    int b = (int)(bl / LSEQ);
    float acc = cb[c];
#pragma unroll
    for (int k = 0; k < KCONV; ++k) {
        int lm = l - (KCONV - 1) + k;
        if (lm >= 0)
            acc += cw[c * KCONV + k] * P[((size_t)b * LSEQ + lm) * PROJ + DIN + c];
    }
    XC[idx] = siluf(acc);
}

// ---------------------------------------------------------------------------
// Sequential SSM scan, head-parallel: one block per (sequence, head),
// 64 blocks total.  Thread (p = tid>>2, nc = tid&3) owns 32 states in VGPRs.
// Emits raw y[b,l,head*64+p] (scan output + D*xh); gating/RMS done later.
// ---------------------------------------------------------------------------
__global__ void __launch_bounds__(256)
scan_kernel(const float* __restrict__ P, const float* __restrict__ XC,
            const float* __restrict__ dtb, const float* __restrict__ Alog,
            const float* __restrict__ Dp, float* __restrict__ Y) {
    __shared__ float sB[NSTATE];
    __shared__ float sC[NSTATE];
    __shared__ float part[DIN];          // part[p*4 + nc]
    int b = blockIdx.x >> 2;
    int head = blockIdx.x & 3;
    int tid = threadIdx.x;
    int p = tid >> 2, nc = tid & 3;
    int nbase = nc * 32;
    float aexp = expf(Alog[head]);
    float dbias = dtb[head];
    float Dh = Dp[head];
    float s[32];
#pragma unroll
    for (int n = 0; n < 32; ++n) s[n] = 0.f;

    for (int l = 0; l < LSEQ; ++l) {
        size_t base = (size_t)b * LSEQ + l;
        __syncthreads();                 // protect sB/sC/part reuse
        if (tid < NSTATE) sB[tid] = XC[base * CONVD + DIN + tid];
        else              sC[tid - NSTATE] = XC[base * CONVD + DIN + NSTATE + (tid - NSTATE)];
        __syncthreads();
        float xh  = XC[base * CONVD + head * HD + p];
        float dtr = P[base * PROJ + DIN + CONVD + head] + dbias;
        float dtv = dtr > 20.f ? dtr : log1pf(expf(dtr));   // softplus
        float dA  = expf(-dtv * aexp);
        float dtx = dtv * xh;
        float acc = 0.f;
#pragma unroll
        for (int n = 0; n < 32; ++n) {
            s[n] = dA * s[n] + dtx * sB[nbase + n];
            acc += s[n] * sC[nbase + n];
        }
        part[tid] = acc;
        __syncthreads();
        if (nc == 0) {
            float y = part[tid] + part[tid + 1] + part[tid + 2] + part[tid + 3] + Dh * xh;
            Y[base * DIN + head * HD + p] = y;
        }
    }
}

// ---------------------------------------------------------------------------
// Gate + group RMS norm: g = rms_over_256(y * silu(z)) * gnorm_w.
// One wave32 per (b,l) row, 8 channels per lane.
// ---------------------------------------------------------------------------
__global__ void gate_kernel(const float* __restrict__ Y,
                            const float* __restrict__ P,
                            const float* __restrict__ gwv,
                            float* __restrict__ G, int rows) {
    int row = blockIdx.x * 8 + (threadIdx.x >> 5);
    if (row >= rows) return;
    int lane = threadIdx.x & 31;
    const float* yrow = Y + (size_t)row * DIN;
    const float* zrow = P + (size_t)row * PROJ;
    float v[8];
    float ss = 0.f;
#pragma unroll
    for (int i = 0; i < 8; ++i) {
        int d = lane * 8 + i;
        float val = yrow[d] * siluf(zrow[d]);
        v[i] = val;
        ss += val * val;
    }
    ss = wave_sum32(ss);
    float sc = rsqrtf(ss / (float)DIN + EPSF);
#pragma unroll
    for (int i = 0; i < 8; ++i) {
        int d = lane * 8 + i;
        G[(size_t)row * DIN + d] = v[i] * sc * gwv[d];
    }
}

// ---------------------------------------------------------------------------
// Out-projection GEMM with fused residual:  H[m,n] += G[m,:] . out_w[n,:] + b
// K = 256, N = 128 (8 tiles).  A-tile staged to LDS (async).
// ---------------------------------------------------------------------------
__global__ void outproj_kernel(const float* __restrict__ Gm,
                               const float* __restrict__ W,
                               const float* __restrict__ bias,
                               float* __restrict__ H) {
    __shared__ float lA[16 * DIN];
    int wave = threadIdx.x >> 5, lane = threadIdx.x & 31;
    int hf = lane >> 4, ln = lane & 15;
    int rowb = blockIdx.x * 16;
    stage_tile<DIN>(Gm + (size_t)rowb * DIN, lA, threadIdx.x);

    int ntile = blockIdx.y * 4 + wave;
    int ncol = ntile * 16 + ln;
    const float* Arow = lA + ln * DIN;
    const float* Wrow = W + (size_t)ncol * DIN;
    v8f acc = wmma_tile_f32(Arow, Wrow, DIN, hf);
    float bv = bias[ncol];
#pragma unroll
    for (int r = 0; r < 8; ++r) {
        int row = rowb + r + 8 * hf;
        H[(size_t)row * HID + ncol] += acc[r] + bv;
    }
}

// ---------------------------------------------------------------------------
// Final RMS + bidirectional merge: m[b,l] = 0.5*(rms(h[b,l]) + rms(h[b+8,L-1-l]))
// ---------------------------------------------------------------------------
__global__ void combine_kernel(const float* __restrict__ h,
                               const float* __restrict__ fw,
                               float* __restrict__ m) {
    int row = blockIdx.x * 8 + (threadIdx.x >> 5);
    if (row >= B0 * LSEQ) return;
    int lane = threadIdx.x & 31;
    int b = row / LSEQ, l = row % LSEQ;
    const float4* p1 = (const float4*)(h + ((size_t)b * LSEQ + l) * HID);
    const float4* p2 = (const float4*)(h + ((size_t)(b + B0) * LSEQ + (LSEQ - 1 - l)) * HID);
    float4 v1 = p1[lane], v2 = p2[lane];
    float s1 = wave_sum32(v1.x * v1.x + v1.y * v1.y + v1.z * v1.z + v1.w * v1.w);
    float s2 = wave_sum32(v2.x * v2.x + v2.y * v2.y + v2.z * v2.z + v2.w * v2.w);
    float c1 = rsqrtf(s1 / (float)HID + EPSF);
    float c2 = rsqrtf(s2 / (float)HID + EPSF);
    float4 wv = ((const float4*)fw)[lane];
    float4 o;
    o.x = 0.5f * (v1.x * c1 + v2.x * c2) * wv.x;
    o.y = 0.5f * (v1.y * c1 + v2.y * c2) * wv.y;
    o.z = 0.5f * (v1.z * c1 + v2.z * c2) * wv.z;
    o.w = 0.5f * (v1.w * c1 + v2.w * c2) * wv.w;
    ((float4*)(m + (size_t)row * HID))[lane] = o;
}

// ---------------------------------------------------------------------------
// MLP layer 1: t1 = gelu(m @ r1_w^T + r1_b),  K = N = 128.
// ---------------------------------------------------------------------------
__global__ void mlp1_kernel(const float* __restrict__ M,
                            const float* __restrict__ W,
                            const float* __restrict__ bias,
                            float* __restrict__ T1) {
    __shared__ float lA[16 * HID];
    int wave = threadIdx.x >> 5, lane = threadIdx.x & 31;
    int hf = lane >> 4, ln = lane & 15;
    int rowb = blockIdx.x * 16;
    stage_tile<HID>(M + (size_t)rowb * HID, lA, threadIdx.x);

    int ntile = blockIdx.y * 4 + wave;
    int ncol = ntile * 16 + ln;
    const float* Arow = lA + ln * HID;
    const float* Wrow = W + (size_t)ncol * HID;
    v8f acc = wmma_tile_f32(Arow, Wrow, HID, hf);
    float bv = bias[ncol];
#pragma unroll
    for (int r = 0; r < 8; ++r) {
        int row = rowb + r + 8 * hf;
        float v = acc[r] + bv;
        T1[(size_t)row * HID + ncol] = 0.5f * v * (1.f + erff(v * 0.70710678f)); // exact gelu
    }
}

// ---------------------------------------------------------------------------
// MLP layer 2 + scatter to (B, CIN, HIMG, WIMG):  N = 480 (30 tiles), K = 128.
// col e -> (p, q, c) = (e/24, (e%24)/3, e%3); row -> (b, i, j).
// ---------------------------------------------------------------------------
__global__ void mlp2_kernel(const float* __restrict__ T1,
                            const float* __restrict__ W,
                            const float* __restrict__ bias,
                            float* __restrict__ out) {
    __shared__ float lA[16 * HID];
    int wave = threadIdx.x >> 5, lane = threadIdx.x & 31;
    int hf = lane >> 4, ln = lane & 15;
    int rowb = blockIdx.x * 16;
    stage_tile<HID>(T1 + (size_t)rowb * HID, lA, threadIdx.x);

    int ntile = blockIdx.y * 4 + wave;
    if (ntile < 30) {                    // wave-uniform
        int e = ntile * 16 + ln;         // 0..479, always valid
        const float* Arow = lA + ln * HID;
        const float* Wrow = W + (size_t)e * HID;
        v8f acc = wmma_tile_f32(Arow, Wrow, HID, hf);
        float bv = bias[e];
        int p = e / (PW * CIN);
        int rem = e % (PW * CIN);
        int q = rem / CIN, c = rem % CIN;
#pragma unroll
        for (int r = 0; r < 8; ++r) {
            int row = rowb + r + 8 * hf;
            int b = row / LSEQ, l = row % LSEQ;
            int i = l / GW, j = l % GW;
            out[(((size_t)b * CIN + c) * HIMG + i * PH + p) * WIMG + j * PW + q] = acc[r] + bv;
        }
    }
}

// ---------------------------------------------------------------------------
// Host-side launch
// ---------------------------------------------------------------------------
extern "C" void kernel_launch(void* const* d_in, const int* in_sizes, int n_in,
                              void* d_out, int out_size, void* d_ws, size_t ws_size,
                              hipStream_t stream) {
    const float* x       = (const float*)d_in[0];
    const float* patch_w = (const float*)d_in[1];
    const float* patch_b = (const float*)d_in[2];
    const float* ln_w    = (const float*)d_in[3];
    const float* in_w    = (const float*)d_in[4];
    const float* in_b    = (const float*)d_in[5];
    const float* conv_w  = (const float*)d_in[6];
    const float* conv_b  = (const float*)d_in[7];
    const float* dt_bias = (const float*)d_in[8];
    const float* A_log   = (const float*)d_in[9];
    const float* Dp      = (const float*)d_in[10];
    const float* gnorm_w = (const float*)d_in[11];
    const float* out_w   = (const float*)d_in[12];
    const float* out_b   = (const float*)d_in[13];
    const float* fnorm_w = (const float*)d_in[14];
    const float* r1_w    = (const float*)d_in[15];
    const float* r1_b    = (const float*)d_in[16];
    const float* r2_w    = (const float*)d_in[17];
    const float* r2_b    = (const float*)d_in[18];

    char* ws = (char*)d_ws;
    size_t off = 0;
    auto carve = [&](size_t elems) {
        float* p = (float*)(ws + off);
        off = (off + elems * sizeof(float) + 255) & ~(size_t)255;
        return p;
    };
    float* h    = carve((size_t)BT * LSEQ * HID);
    float* u    = carve((size_t)BT * LSEQ * HID);
    float* proj = carve((size_t)BT * LSEQ * PROJ);
    float* xc   = carve((size_t)BT * LSEQ * CONVD);
    float* ybuf = carve((size_t)BT * LSEQ * DIN);
    float* gbuf = carve((size_t)BT * LSEQ * DIN);
    float* mbuf = carve((size_t)B0 * LSEQ * HID);
    float* t1   = carve((size_t)B0 * LSEQ * HID);
    (void)ws_size; (void)in_sizes; (void)n_in; (void)out_size;

    patch_kernel<<<B0 * LSEQ, 128, 0, stream>>>(x, patch_w, patch_b, h);

    for (int i = 0; i < NL; ++i) {
        rms_kernel<<<(BT * LSEQ + 7) / 8, 256, 0, stream>>>(h, ln_w + i * HID, u, BT * LSEQ);

        dim3 g1(BT * LSEQ / 16, ((PROJ + 15) / 16 + 3) / 4); // 1530 x 13
        inproj_kernel<<<g1, 128, 0, stream>>>(u, in_w + (size_t)i * PROJ * HID,
                                              in_b + i * PROJ, proj);

        size_t ce = (size_t)BT * LSEQ * CONVD;
        conv_kernel<<<(unsigned)((ce + 255) / 256), 256, 0, stream>>>(
            proj, conv_w + i * CONVD * KCONV, conv_b + i * CONVD, xc);

        scan_kernel<<<BT * NH, 256, 0, stream>>>(proj, xc, dt_bias + i * NH,
                                                 A_log + i * NH, Dp + i * NH, ybuf);

        gate_kernel<<<(BT * LSEQ + 7) / 8, 256, 0, stream>>>(ybuf, proj,
                                                             gnorm_w + i * DIN,
                                                             gbuf, BT * LSEQ);

        dim3 g2(BT * LSEQ / 16, 2);                          // 8 col tiles / 4 waves
        outproj_kernel<<<g2, 128, 0, stream>>>(gbuf, out_w + (size_t)i * HID * DIN,
                                               out_b + i * HID, h);
    }

    combine_kernel<<<(B0 * LSEQ + 7) / 8, 256, 0, stream>>>(h, fnorm_w, mbuf);

    dim3 g3(B0 * LSEQ / 16, 2);
    mlp1_kernel<<<g3, 128, 0, stream>>>(mbuf, r1_w, r1_b, t1);

    dim3 g4(B0 * LSEQ / 16, 8);                              // 30 col tiles
    mlp2_kernel<<<g4, 128, 0, stream>>>(t1, r2_w, r2_b, (float*)d_out);
}